// HeteroRGCNLayer_11716670783786
// MI455X (gfx1250) — compile-verified
//
#include <hip/hip_runtime.h>
#include <hip/hip_bf16.h>

#define N_U 50000
#define N_V 50000
#define NEDGE 600000
#define DIM 128

typedef __attribute__((ext_vector_type(2))) float v2f;
typedef __attribute__((ext_vector_type(8))) float v8f;

// ---------------------------------------------------------------------------
// Zero helpers (graph-capture safe; avoid hipMemsetAsync edge cases)
// ---------------------------------------------------------------------------
__global__ __launch_bounds__(256) void zero_f32_kernel(float* __restrict__ p, int n4) {
    int i = blockIdx.x * 256 + threadIdx.x;
    if (i < n4) {
        float4 z = make_float4(0.f, 0.f, 0.f, 0.f);
        ((float4*)p)[i] = z;
    }
}

__global__ __launch_bounds__(256) void zero_i32_kernel(int* __restrict__ p, int n) {
    int i = blockIdx.x * 256 + threadIdx.x;
    if (i < n) p[i] = 0;
}

// ---------------------------------------------------------------------------
// Degree count: cnt[dst[e]] += 1
// ---------------------------------------------------------------------------
__global__ __launch_bounds__(256) void count_kernel(const int* __restrict__ dst,
                                                    int* __restrict__ cnt, int n) {
    int i = blockIdx.x * 256 + threadIdx.x;
    if (i < n) atomicAdd(&cnt[dst[i]], 1);
}

// ---------------------------------------------------------------------------
// Wh = A @ W^T + bias via V_WMMA_F32_16X16X4_F32 (full f32 precision).
// One wave per 16-row strip; computes all 8 column tiles (N=128), K=128.
//
// A frag (16x4 f32, 2 VGPRs): lane L -> row (L&15), a = A[row][4k + (L>>4)*2 .. +1]
// B frag (4x16 f32, 2 VGPRs): lane L -> col (L&15); B[k][n] = W[n][k]
//                             b = W[col][4k + (L>>4)*2 .. +1]  (contiguous float2)
// C/D (16x16 f32, 8 VGPRs):   vgpr r, lane L -> row r + (L>>4)*8, col (L&15)
// ---------------------------------------------------------------------------
__global__ __launch_bounds__(32) void gemm_bias_wmma_kernel(
    const float* __restrict__ A,     // [M, 128] row-major
    const float* __restrict__ W,     // [128, 128] row-major (D_OUT, D_IN)
    const float* __restrict__ bias,  // [128]
    float* __restrict__ C,           // [M, 128]
    int mtiles)
{
    int mtile = blockIdx.x;
    if (mtile >= mtiles) return;
    int lane   = threadIdx.x;        // 0..31 (wave32)
    int l15    = lane & 15;
    int khalf  = (lane >> 4) * 2;    // 0 or 2

    const float* arow = A + (size_t)(mtile * 16 + l15) * DIM;

    v8f acc0 = {0.f,0.f,0.f,0.f,0.f,0.f,0.f,0.f};
    v8f acc[8];
    #pragma unroll
    for (int n = 0; n < 8; ++n) acc[n] = acc0;

    for (int k = 0; k < DIM / 4; ++k) {
        int kb = k * 4 + khalf;
        v2f a = *(const v2f*)(arow + kb);          // 8B aligned
        #pragma unroll
        for (int n = 0; n < 8; ++n) {
            int col = n * 16 + l15;
            v2f b = *(const v2f*)(W + (size_t)col * DIM + kb);
            acc[n] = __builtin_amdgcn_wmma_f32_16x16x4_f32(
                /*neg_a=*/false, a, /*neg_b=*/false, b,
                /*c_mod=*/(short)0, acc[n],
                /*reuse_a=*/false, /*reuse_b=*/false);
        }
    }

    int rowhi = (lane >> 4) * 8;
    #pragma unroll
    for (int n = 0; n < 8; ++n) {
        int col = n * 16 + l15;
        float bv = bias[col];
        #pragma unroll
        for (int r = 0; r < 8; ++r) {
            int row = mtile * 16 + rowhi + r;
            C[(size_t)row * DIM + col] = acc[n][r] + bv;
        }
    }
}

// ---------------------------------------------------------------------------
// Scaled scatter: out[dst[e]] += Wh[src[e]] * (1 / max(cnt[dst[e]], 1))
// One wave (32 lanes) per edge; each lane moves a float4 (128 floats total).
// mean = sum(msg)/cnt == sum(msg/cnt), so this accumulates the segment-mean
// (summed across etypes onto the same dst ntype) directly into d_out.
// ---------------------------------------------------------------------------
__global__ __launch_bounds__(256) void scatter_mean_kernel(
    const float* __restrict__ Wh,    // [N_src, 128]
    const int* __restrict__ src,
    const int* __restrict__ dst,
    const int* __restrict__ cnt,     // [n_dst] in-degrees for this etype
    float* __restrict__ out,         // [n_dst, 128] accumulator (pre-zeroed)
    int nedges)
{
    int tid  = blockIdx.x * 256 + threadIdx.x;
    int e    = tid >> 5;
    int lane = tid & 31;
    if (e >= nedges) return;

    int s = src[e];
    int d = dst[e];
    int c = cnt[d];
    float inv = 1.0f / (float)(c > 1 ? c : 1);

    const float4 v = *(const float4*)(Wh + (size_t)s * DIM + lane * 4);
    float* o = out + (size_t)d * DIM + lane * 4;
    atomicAdd(o + 0, v.x * inv);
    atomicAdd(o + 1, v.y * inv);
    atomicAdd(o + 2, v.z * inv);
    atomicAdd(o + 3, v.w * inv);
}

// ---------------------------------------------------------------------------
// Launcher
// ---------------------------------------------------------------------------
extern "C" void kernel_launch(void* const* d_in, const int* in_sizes, int n_in,
                              void* d_out, int out_size, void* d_ws, size_t ws_size,
                              hipStream_t stream) {
    (void)in_sizes; (void)n_in; (void)out_size; (void)ws_size;

    const float* feat_u = (const float*)d_in[0];
    const float* feat_v = (const float*)d_in[1];
    const float* W_uu   = (const float*)d_in[2];
    const float* b_uu   = (const float*)d_in[3];
    const float* W_uv   = (const float*)d_in[4];
    const float* b_uv   = (const float*)d_in[5];
    const float* W_vu   = (const float*)d_in[6];
    const float* b_vu   = (const float*)d_in[7];
    const int* src_uu   = (const int*)d_in[8];
    const int* dst_uu   = (const int*)d_in[9];
    const int* src_uv   = (const int*)d_in[10];
    const int* dst_uv   = (const int*)d_in[11];
    const int* src_vu   = (const int*)d_in[12];
    const int* dst_vu   = (const int*)d_in[13];

    float* out   = (float*)d_out;                    // [2, 50000, 128]
    float* out_u = out;
    float* out_v = out + (size_t)N_U * DIM;

    // Workspace layout: 3x Wh [50000,128] f32, then 3x cnt [50000] i32.
    float* Wh_uu = (float*)d_ws;
    float* Wh_uv = Wh_uu + (size_t)N_U * DIM;
    float* Wh_vu = Wh_uv + (size_t)N_U * DIM;
    int*   cnt_uu = (int*)(Wh_vu + (size_t)N_V * DIM);
    int*   cnt_uv = cnt_uu + N_U;
    int*   cnt_vu = cnt_uv + N_V;

    // 1) zero output and counters
    {
        int n4 = (2 * N_U * DIM) / 4;                // 3.2M float4
        zero_f32_kernel<<<(n4 + 255) / 256, 256, 0, stream>>>(out, n4);
        int nc = N_U + N_V + N_U;                    // 150000 ints
        zero_i32_kernel<<<(nc + 255) / 256, 256, 0, stream>>>(cnt_uu, nc);
    }

    // 2) per-etype in-degree counts
    {
        int blocks = (NEDGE + 255) / 256;
        count_kernel<<<blocks, 256, 0, stream>>>(dst_uu, cnt_uu, NEDGE);
        count_kernel<<<blocks, 256, 0, stream>>>(dst_uv, cnt_uv, NEDGE);
        count_kernel<<<blocks, 256, 0, stream>>>(dst_vu, cnt_vu, NEDGE);
    }

    // 3) WMMA GEMMs: Wh_x = feat_src @ W_x^T + b_x
    {
        int mtiles = N_U / 16;                       // 3125, exact
        gemm_bias_wmma_kernel<<<mtiles, 32, 0, stream>>>(feat_u, W_uu, b_uu, Wh_uu, mtiles);
        gemm_bias_wmma_kernel<<<mtiles, 32, 0, stream>>>(feat_u, W_uv, b_uv, Wh_uv, mtiles);
        gemm_bias_wmma_kernel<<<mtiles, 32, 0, stream>>>(feat_v, W_vu, b_vu, Wh_vu, mtiles);
    }

    // 4) scaled atomic scatter of segment-means directly into d_out
    {
        int blocks = (NEDGE * 32 + 255) / 256;       // 75000
        scatter_mean_kernel<<<blocks, 256, 0, stream>>>(Wh_uu, src_uu, dst_uu, cnt_uu, out_u, NEDGE);
        scatter_mean_kernel<<<blocks, 256, 0, stream>>>(Wh_vu, src_vu, dst_vu, cnt_vu, out_u, NEDGE);
        scatter_mean_kernel<<<blocks, 256, 0, stream>>>(Wh_uv, src_uv, dst_uv, cnt_uv, out_v, NEDGE);
    }
}